// ArgmaxMultiCodeDecoder_66709432041870
// MI455X (gfx1250) — compile-verified
//
#include <hip/hip_runtime.h>
#include <hip/hip_bf16.h>
#include <math.h>

#define H 1024
#define NH 16
#define NKV 8
#define HD 128
#define LAYERS 5
#define SEQ 16
#define FF 4096
#define NLM 15
#define VOC 2048
#define KVD (NKV * HD)

typedef __attribute__((ext_vector_type(2))) float v2f;
typedef __attribute__((ext_vector_type(8))) float v8f;
typedef __attribute__((ext_vector_type(4))) unsigned int u32x4;
typedef __attribute__((ext_vector_type(8))) int i32x8;
typedef __attribute__((ext_vector_type(4))) int i32x4;

#define CHUNK_K 256   // K-depth per block (split-K along grid.z)
#define TILE_N  32    // outputs per block (2 waves x 16)
#define SUB_K   64    // rows per TDM sub-tile
#define NSUB    (CHUNK_K / SUB_K)   // 4 double-buffered stages

// ---------------- prep: copy hidden, build cos/sin tables -------------------
__global__ void prep_kernel(const float* __restrict__ emb,
                            const float* __restrict__ cache_len,
                            const float* __restrict__ inv_freq,
                            float* __restrict__ hbuf,
                            float* __restrict__ cs) {
  int t = threadIdx.x;  // 128 threads
  for (int i = t; i < H; i += 128) hbuf[i] = emb[i];
  if (t < HD / 2) {
    float f = inv_freq[t] * cache_len[0];
    float c = cosf(f), s = sinf(f);           // SCALING == 1.0
    cs[t] = c;        cs[t + 64] = c;         // cos over concat(freqs,freqs)
    cs[128 + t] = s;  cs[192 + t] = s;        // sin
  }
}

// ---------------- rmsnorm (one block, 256 threads) --------------------------
__global__ void rmsnorm_kernel(const float* __restrict__ hin,
                               const float* __restrict__ w,
                               float* __restrict__ xout,
                               float* __restrict__ copy_out) {
  __shared__ float sred[256];
  int tid = threadIdx.x;
  float ss = 0.f;
  for (int i = tid; i < H; i += 256) { float v = hin[i]; ss += v * v; }
  sred[tid] = ss; __syncthreads();
  for (int st = 128; st > 0; st >>= 1) {
    if (tid < st) sred[tid] += sred[tid + st];
    __syncthreads();
  }
  if (tid == 0) sred[0] = rsqrtf(sred[0] / (float)H + 1e-6f);
  __syncthreads();
  float sc = sred[0];
  for (int i = tid; i < H; i += 256) {
    xout[i] = hin[i] * sc * w[i];
    if (copy_out) copy_out[i] = hin[i];
  }
}

__global__ void zero_kernel(float* __restrict__ p, int n) {
  int i = blockIdx.x * 256 + threadIdx.x;
  if (i < n) p[i] = 0.f;
}

__global__ void silu_mul_kernel(float* __restrict__ g, const float* __restrict__ u, int n) {
  int i = blockIdx.x * 256 + threadIdx.x;
  if (i < n) { float gv = g[i]; g[i] = (gv / (1.f + expf(-gv))) * u[i]; }
}

// ---------------- GEMV: TDM tile stream -> LDS -> V_WMMA_F32_16X16X4_F32 ----
// out[b][j] += sum_k x[k] * W[b][k*N + j]
// Each wave owns 16 output columns. A (16x4) = x broadcast into all 16 rows
// (from LDS); B (4x16) = row-major W tile, streamed into LDS by the Tensor
// Data Mover as SUB_K x 16 sub-tiles (tensor_dim0_stride = N), double
// buffered and pipelined on TENSORcnt. Result row M=0 is acc[0], lanes 0-15.

__device__ __forceinline__ void tdm_issue_tile(const float* gaddr, unsigned lds_addr,
                                               int N) {
  unsigned long long ga = (unsigned long long)(uintptr_t)gaddr;
  u32x4 g0;
  g0[0] = 1u;                                    // count=1, user mode
  g0[1] = lds_addr;                              // LDS byte address
  g0[2] = (unsigned)ga;                          // global_addr[31:0]
  g0[3] = (unsigned)((ga >> 32) & 0x01FFFFFFu) | (2u << 30);  // addr[56:32] | type=2
  i32x8 g1;
  g1[0] = (int)(2u << 16);        // workgroup_mask=0, data_size=2 (4B)
  g1[1] = (int)((unsigned)N << 16);   // tensor_dim0[15:0] in bits 63:48
  g1[2] = (int)((unsigned)SUB_K << 16); // dim0 hi=0 | tensor_dim1[15:0]=SUB_K
  g1[3] = (int)(16u << 16);       // tensor_dim1 hi=0 | tile_dim0=16
  g1[4] = SUB_K;                  // tile_dim1=SUB_K, tile_dim2=0
  g1[5] = N;                      // tensor_dim0_stride[31:0] = N
  g1[6] = 0;                      // stride hi / dim1_stride lo
  g1[7] = 0;
  i32x4 gz4 = {0, 0, 0, 0};
  i32x8 gz8 = {0, 0, 0, 0, 0, 0, 0, 0};
  __builtin_amdgcn_tensor_load_to_lds(g0, g1, gz4, gz4, gz8, 0);
}

__global__ void gemv_wmma_kernel(const float* __restrict__ x,
                                 const float* __restrict__ W,
                                 float* __restrict__ out,
                                 int N, long strideWb, int strideOut) {
  __shared__ float lx[CHUNK_K];
  __shared__ float tile[2][2][SUB_K * 16];  // [wave][buf][row*16+col]
  const int tid  = threadIdx.x;   // 64 threads = 2 waves
  const int lane = tid & 31;
  const int wave = __builtin_amdgcn_readfirstlane(tid >> 5);  // wave-uniform
  const float* Wb   = W + (long)blockIdx.y * strideWb;
  float*       outb = out + (long)blockIdx.y * strideOut;
  const int k0 = blockIdx.z * CHUNK_K;
  const int j0 = blockIdx.x * TILE_N + wave * 16;
  const int n15  = lane & 15;
  const int jc   = j0 + n15;
  const int roff = (lane >> 4) << 1;  // K-offset: 0 (lanes 0-15) / 2 (lanes 16-31)

  const unsigned ldsA = (unsigned)(uintptr_t)(void*)&tile[wave][0][0];
  const unsigned ldsB = (unsigned)(uintptr_t)(void*)&tile[wave][1][0];

  // kick off first tile DMA, overlap with x-chunk staging
  tdm_issue_tile(Wb + (long)k0 * N + j0, ldsA, N);

  for (int t = tid; t < CHUNK_K; t += 64) lx[t] = x[k0 + t];
  __syncthreads();

  v8f acc0 = {}; v8f acc1 = {};
  for (int s = 0; s < NSUB; ++s) {
    if (s + 1 < NSUB) {
      tdm_issue_tile(Wb + (long)(k0 + (s + 1) * SUB_K) * N + j0,
                     ((s + 1) & 1) ? ldsB : ldsA, N);
      __builtin_amdgcn_s_wait_tensorcnt(1);  // in-order: tile s resident
    } else {
      __builtin_amdgcn_s_wait_tensorcnt(0);
    }
    const float* tb = &tile[wave][s & 1][0];
    const float* la = &lx[s * SUB_K];
#pragma unroll
    for (int kk = 0; kk < SUB_K; kk += 8) {
      v2f a0, b0, a1, b1;
      a0.x = la[kk + roff];       a0.y = la[kk + roff + 1];
      a1.x = la[kk + 4 + roff];   a1.y = la[kk + 5 + roff];
      b0.x = tb[(kk + roff) * 16 + n15];     b0.y = tb[(kk + roff + 1) * 16 + n15];
      b1.x = tb[(kk + 4 + roff) * 16 + n15]; b1.y = tb[(kk + 5 + roff) * 16 + n15];
      acc0 = __builtin_amdgcn_wmma_f32_16x16x4_f32(false, a0, false, b0,
                                                   (short)0, acc0, false, false);
      acc1 = __builtin_amdgcn_wmma_f32_16x16x4_f32(false, a1, false, b1,
                                                   (short)0, acc1, false, false);
    }
  }
  float r = acc0[0] + acc1[0];  // M=0 row, column jc (lanes 0-15)
  if (lane < 16)
    __hip_atomic_fetch_add(&outb[jc], r, __ATOMIC_RELAXED, __HIP_MEMORY_SCOPE_AGENT);
}

// ---------------- attention: rope + cache update + softmax + O --------------
__global__ void attn_kernel(const float* __restrict__ q_in,   // [NH*HD]
                            const float* __restrict__ k_in,   // [NKV*HD]
                            const float* __restrict__ v_in,   // [NKV*HD]
                            const float* __restrict__ kcb,    // key_cache layer base [KVD*SEQ]
                            const float* __restrict__ vcb,    // value_cache layer base
                            const float* __restrict__ pad,    // [SEQ]
                            const float* __restrict__ upd,    // [SEQ]
                            const float* __restrict__ cs,     // cos[128], sin[128]
                            float* __restrict__ attn_o,       // [NH*HD]
                            float* __restrict__ out_key,      // d_out slice [KVD*SEQ]
                            float* __restrict__ out_val) {
  __shared__ float qs[NH * HD];
  __shared__ float ks[NKV * HD];
  __shared__ float vs[NKV * HD];
  __shared__ float sc[NH * SEQ];
  __shared__ float red[NH * 2];
  const int tid = threadIdx.x;  // 256

  for (int e = tid; e < NH * HD; e += 256) {
    int dd = e & (HD - 1);
    float val = q_in[e];
    float rot = (dd < HD / 2) ? -q_in[e + HD / 2] : q_in[e - HD / 2];
    qs[e] = val * cs[dd] + rot * cs[128 + dd];
  }
  for (int e = tid; e < NKV * HD; e += 256) {
    int dd = e & (HD - 1);
    float val = k_in[e];
    float rot = (dd < HD / 2) ? -k_in[e + HD / 2] : k_in[e - HD / 2];
    ks[e] = val * cs[dd] + rot * cs[128 + dd];
    vs[e] = v_in[e];
  }
  __syncthreads();

  // masked KV-cache update -> outputs
  for (int e = tid; e < KVD * SEQ; e += 256) {
    int r = e >> 4, s = e & (SEQ - 1);
    float mm = upd[s], om = 1.f - mm;
    out_key[e] = kcb[e] * om + ks[r] * mm;
    out_val[e] = vcb[e] * om + vs[r] * mm;
  }

  // scores: one thread per (head, pos)
  {
    int hh = tid >> 4, s = tid & (SEQ - 1);
    int kv = hh >> 1;  // NREP = 2
    float mm = upd[s], om = 1.f - mm;
    const float* kcrow = kcb + (kv * HD) * SEQ + s;
    const float* qrow  = qs + hh * HD;
    const float* krow  = ks + kv * HD;
    float acc = 0.f;
    for (int d = 0; d < HD; ++d)
      acc += qrow[d] * (kcrow[d * SEQ] * om + krow[d] * mm);
    sc[tid] = acc * 0.08838834764831845f + pad[s];  // 1/sqrt(128)
  }
  __syncthreads();

  if (tid < NH) {
    float mx = -1e30f;
    for (int s = 0; s < SEQ; ++s) mx = fmaxf(mx, sc[tid * SEQ + s]);
    float sum = 0.f;
    for (int s = 0; s < SEQ; ++s) sum += expf(sc[tid * SEQ + s] - mx);
    red[tid * 2] = mx; red[tid * 2 + 1] = 1.f / sum;
  }
  __syncthreads();
  {
    int hh = tid >> 4;
    sc[tid] = expf(sc[tid] - red[hh * 2]) * red[hh * 2 + 1];
  }
  __syncthreads();

  // O = probs @ V
  for (int e = tid; e < NH * HD; e += 256) {
    int hh = e >> 7, d = e & (HD - 1);
    int kv = hh >> 1;
    const float* vcrow = vcb + (kv * HD + d) * SEQ;
    float vl = vs[kv * HD + d];
    float acc = 0.f;
    for (int s = 0; s < SEQ; ++s) {
      float mm = upd[s];
      acc += sc[hh * SEQ + s] * (vcrow[s] * (1.f - mm) + vl * mm);
    }
    attn_o[e] = acc;
  }
}

// ---------------------------------------------------------------------------
extern "C" void kernel_launch(void* const* d_in, const int* in_sizes, int n_in,
                              void* d_out, int out_size, void* d_ws, size_t ws_size,
                              hipStream_t stream) {
  // setup_inputs() dict order:
  const float* input_embeds = (const float*)d_in[0];
  const float* cache_length = (const float*)d_in[1];
  const float* key_cache    = (const float*)d_in[2];
  const float* value_cache  = (const float*)d_in[3];
  const float* pad_mask     = (const float*)d_in[4];
  const float* upd_mask     = (const float*)d_in[5];
  const float* Wq = (const float*)d_in[6];
  const float* Wk = (const float*)d_in[7];
  const float* Wv = (const float*)d_in[8];
  const float* Wo = (const float*)d_in[9];
  const float* Wg = (const float*)d_in[10];
  const float* Wu = (const float*)d_in[11];
  const float* Wd = (const float*)d_in[12];
  const float* ln1 = (const float*)d_in[13];
  const float* ln2 = (const float*)d_in[14];
  const float* norm_w = (const float*)d_in[15];
  const float* lm_w = (const float*)d_in[16];
  const float* inv_freq = (const float*)d_in[17];

  float* out = (float*)d_out;
  float* logits = out;                 // [15*2048]
  float* hid    = out + 30720;         // [1024]
  float* nkey   = out + 31744;         // [5120*16]
  float* nval   = out + 113664;        // [5120*16]

  float* ws = (float*)d_ws;
  float* h  = ws;            // 1024
  float* x  = ws + 1024;     // 1024
  float* q  = ws + 2048;     // 2048
  float* k  = ws + 4096;     // 1024
  float* v  = ws + 5120;     // 1024
  float* ao = ws + 6144;     // 2048
  float* g  = ws + 8192;     // 4096
  float* u  = ws + 12288;    // 4096
  float* cs = ws + 16384;    // 256

  auto gemv = [&](const float* xv, const float* Wm, float* ov, int K, int N,
                  int batch, long sWb, int sOut) {
    dim3 grid(N / TILE_N, batch, K / CHUNK_K);
    gemv_wmma_kernel<<<grid, 64, 0, stream>>>(xv, Wm, ov, N, sWb, sOut);
  };
  auto zero = [&](float* p, int n) {
    zero_kernel<<<(n + 255) / 256, 256, 0, stream>>>(p, n);
  };

  prep_kernel<<<1, 128, 0, stream>>>(input_embeds, cache_length, inv_freq, h, cs);

  for (int i = 0; i < LAYERS; ++i) {
    const float* Wq_i = Wq + (size_t)i * H * (NH * HD);
    const float* Wk_i = Wk + (size_t)i * H * KVD;
    const float* Wv_i = Wv + (size_t)i * H * KVD;
    const float* Wo_i = Wo + (size_t)i * (NH * HD) * H;
    const float* Wg_i = Wg + (size_t)i * H * FF;
    const float* Wu_i = Wu + (size_t)i * H * FF;
    const float* Wd_i = Wd + (size_t)i * FF * H;

    rmsnorm_kernel<<<1, 256, 0, stream>>>(h, ln1 + i * H, x, nullptr);

    zero(q, NH * HD);  gemv(x, Wq_i, q, H, NH * HD, 1, 0, 0);
    zero(k, KVD);      gemv(x, Wk_i, k, H, KVD, 1, 0, 0);
    zero(v, KVD);      gemv(x, Wv_i, v, H, KVD, 1, 0, 0);

    attn_kernel<<<1, 256, 0, stream>>>(
        q, k, v,
        key_cache + (size_t)i * KVD * SEQ, value_cache + (size_t)i * KVD * SEQ,
        pad_mask, upd_mask, cs, ao,
        nkey + (size_t)i * KVD * SEQ, nval + (size_t)i * KVD * SEQ);

    // h += ao @ Wo  (accumulate directly into h via atomics)
    gemv(ao, Wo_i, h, NH * HD, H, 1, 0, 0);

    rmsnorm_kernel<<<1, 256, 0, stream>>>(h, ln2 + i * H, x, nullptr);

    zero(g, FF);  gemv(x, Wg_i, g, H, FF, 1, 0, 0);
    zero(u, FF);  gemv(x, Wu_i, u, H, FF, 1, 0, 0);
    silu_mul_kernel<<<FF / 256, 256, 0, stream>>>(g, u, FF);

    // h += (silu(g)*u) @ Wd
    gemv(g, Wd_i, h, FF, H, 1, 0, 0);
  }

  // final norm, copy hidden out, LM heads
  rmsnorm_kernel<<<1, 256, 0, stream>>>(h, norm_w, x, hid);
  zero(logits, NLM * VOC);
  gemv(x, lm_w, logits, H, VOC, NLM, (long)H * VOC, VOC);
}